// ViSNeRF_Original_5196910428445
// MI455X (gfx1250) — compile-verified
//
#include <hip/hip_runtime.h>

typedef __attribute__((ext_vector_type(2))) float v2f;
typedef __attribute__((ext_vector_type(8))) float v8f;

#define RES     512
#define NCOMP   16
#define BOUNDS  1.3f
#define FSTRIDE 21   // padded per-point feature stride in LDS (20 used)
#define OUTD    32
#define KIN     19

__global__ __launch_bounds__(256) void visnerf_wmma_kernel(
    const float* __restrict__ pts,       // [N,3]
    const float* __restrict__ tstamps,   // [N,1]
    const float* __restrict__ planes,    // [3,16,512,512]
    const float* __restrict__ lines,     // [3,16,512]
    const float* __restrict__ pgrid,     // [3,3]
    const float* __restrict__ W,         // [32,19] row-major
    float* __restrict__ out)             // [N,32]
{
    __shared__ float lds[8][32 * FSTRIDE];

    const int tid  = threadIdx.x;
    const int wave = tid >> 5;
    const int lane = tid & 31;
    const int pt   = blockIdx.x * 256 + tid;

    // ---------------- per-lane feature computation ----------------
    // pts_n = (p - B) * (2 / (-2B)) - 1  ==  -p / B
    const float n0 = pts[pt * 3 + 0] * (-1.0f / BOUNDS);
    const float n1 = pts[pt * 3 + 1] * (-1.0f / BOUNDS);
    const float n2 = pts[pt * 3 + 2] * (-1.0f / BOUNDS);
    const float tt = tstamps[pt];

    // MAT_X={0,0,1}, MAT_Y={1,2,2}, VEC_M={2,1,0}
    const float px[3] = {n0, n0, n1};
    const float py[3] = {n1, n2, n2};
    const float lc[3] = {n2, n1, n0};

    float coef[NCOMP];
#pragma unroll
    for (int c = 0; c < NCOMP; ++c) coef[c] = 0.0f;

#pragma unroll
    for (int i = 0; i < 3; ++i) {
        // bilinear setup (align_corners=True on a 512x512 grid)
        const float ix = (px[i] + 1.0f) * 0.5f * (float)(RES - 1);
        const float iy = (py[i] + 1.0f) * 0.5f * (float)(RES - 1);
        const float fx = floorf(ix), fy = floorf(iy);
        const float wx = ix - fx,    wy = iy - fy;
        const int x0 = min(max((int)fx, 0), RES - 1);
        const int x1 = min(x0 + 1, RES - 1);
        const int y0 = min(max((int)fy, 0), RES - 1);
        const int y1 = min(y0 + 1, RES - 1);
        const float w00 = (1.0f - wy) * (1.0f - wx);
        const float w01 = (1.0f - wy) * wx;
        const float w10 = wy * (1.0f - wx);
        const float w11 = wy * wx;

        // 1D line interp setup
        const float ly = (lc[i] + 1.0f) * 0.5f * (float)(RES - 1);
        const float fl = floorf(ly);
        const float wl = ly - fl;
        const int l0 = min(max((int)fl, 0), RES - 1);
        const int l1 = min(l0 + 1, RES - 1);

        const float* __restrict__ pbase = planes + (size_t)i * NCOMP * RES * RES;
        const float* __restrict__ lbase = lines  + (size_t)i * NCOMP * RES;

#pragma unroll
        for (int c = 0; c < NCOMP; ++c) {
            const float* __restrict__ pg = pbase + (size_t)c * RES * RES;
            const float v00 = pg[y0 * RES + x0];
            const float v01 = pg[y0 * RES + x1];
            const float v10 = pg[y1 * RES + x0];
            const float v11 = pg[y1 * RES + x1];
            const float pf  = v00 * w00 + v01 * w01 + v10 * w10 + v11 * w11;
            const float* __restrict__ lg = lbase + c * RES;
            const float lf = lg[l0] * (1.0f - wl) + lg[l1] * wl;
            coef[c] = fmaf(pf, lf, coef[c]);
        }
    }

    // params_grid [3,3], linear interp along length-3 axis
    const float ty = (tt + 1.0f) * 0.5f * 2.0f;
    const float ft = floorf(ty);
    const float wt = ty - ft;
    const int t0 = min(max((int)ft, 0), 2);
    const int t1 = min(t0 + 1, 2);
    const float pf0 = pgrid[0 * 3 + t0] * (1.0f - wt) + pgrid[0 * 3 + t1] * wt;
    const float pf1 = pgrid[1 * 3 + t0] * (1.0f - wt) + pgrid[1 * 3 + t1] * wt;
    const float pf2 = pgrid[2 * 3 + t0] * (1.0f - wt) + pgrid[2 * 3 + t1] * wt;

    // stage feats[20] (K padded to 20 with a trailing zero) to LDS
    float* __restrict__ f = &lds[wave][lane * FSTRIDE];
#pragma unroll
    for (int c = 0; c < NCOMP; ++c) f[c] = coef[c];
    f[16] = pf0; f[17] = pf1; f[18] = pf2; f[19] = 0.0f;

    __syncthreads();

    // ---------------- B-matrix (W) tiles into registers ----------------
    // B[k][o] = W[o][k]. ISA 32-bit B layout (4x16): VGPR0/1 hold K={0,1} for
    // lanes 0-15 (N=lane) and K={2,3} for lanes 16-31 (N=lane-16).
    const int ohalf = lane & 15;
    const int khi   = (lane >> 4) << 1;  // 0 or 2

    v2f Breg[2][5];
#pragma unroll
    for (int nt = 0; nt < 2; ++nt) {
        const int o = nt * 16 + ohalf;
#pragma unroll
        for (int s = 0; s < 5; ++s) {
            const int k0 = 4 * s + khi;
            const float b0 = W[o * KIN + min(k0, KIN - 1)];
            const float b1 = W[o * KIN + min(k0 + 1, KIN - 1)];
            Breg[nt][s].x = (k0     < KIN) ? b0 : 0.0f;
            Breg[nt][s].y = (k0 + 1 < KIN) ? b1 : 0.0f;
        }
    }

    // ---------------- WMMA: feats[32x20] @ B[20x32] ----------------
    const float* __restrict__ fb = &lds[wave][0];
    v8f acc[2][2];
#pragma unroll
    for (int g = 0; g < 2; ++g)
#pragma unroll
        for (int nt = 0; nt < 2; ++nt)
#pragma unroll
            for (int v = 0; v < 8; ++v) acc[g][nt][v] = 0.0f;

    const int m = lane & 15;
#pragma unroll
    for (int g = 0; g < 2; ++g) {          // two 16-point A tiles per wave
#pragma unroll
        for (int s = 0; s < 5; ++s) {      // K = 20 in steps of 4
            const int kb = 4 * s + khi;
            v2f a;
            a.x = fb[(g * 16 + m) * FSTRIDE + kb];
            a.y = fb[(g * 16 + m) * FSTRIDE + kb + 1];
            acc[g][0] = __builtin_amdgcn_wmma_f32_16x16x4_f32(
                false, a, false, Breg[0][s], (short)0, acc[g][0], false, false);
            acc[g][1] = __builtin_amdgcn_wmma_f32_16x16x4_f32(
                false, a, false, Breg[1][s], (short)0, acc[g][1], false, false);
        }
    }

    // ---------------- store D tiles ----------------
    // C/D layout: VGPR v holds row m=v (lanes 0-15, col=lane) / m=v+8 (lanes 16-31).
    const int pbase = blockIdx.x * 256 + wave * 32;
    const int mhi   = (lane >> 4) * 8;
#pragma unroll
    for (int g = 0; g < 2; ++g) {
#pragma unroll
        for (int nt = 0; nt < 2; ++nt) {
            const int col = nt * 16 + ohalf;
#pragma unroll
            for (int v = 0; v < 8; ++v) {
                const int row = pbase + g * 16 + mhi + v;
                out[(size_t)row * OUTD + col] = acc[g][nt][v];
            }
        }
    }
}

extern "C" void kernel_launch(void* const* d_in, const int* in_sizes, int n_in,
                              void* d_out, int out_size, void* d_ws, size_t ws_size,
                              hipStream_t stream) {
    const float* pts     = (const float*)d_in[0];
    const float* tstamps = (const float*)d_in[1];
    const float* planes  = (const float*)d_in[2];
    const float* lines   = (const float*)d_in[3];
    const float* pgrid   = (const float*)d_in[4];
    const float* W       = (const float*)d_in[5];
    float* out           = (float*)d_out;

    const int N = in_sizes[0] / 3;        // 2,097,152
    const int blocks = N / 256;           // exact: N is a multiple of 256

    visnerf_wmma_kernel<<<blocks, 256, 0, stream>>>(
        pts, tstamps, planes, lines, pgrid, W, out);
}